// Dense3DPointsToRenderedSubPixelDepth_9783935500913
// MI455X (gfx1250) — compile-verified
//
#include <hip/hip_runtime.h>
#include <stdint.h>

// Z-buffer point splatting for MI455X (gfx1250, wave32).
//
// Pass 1 (init):    ws[pixel] = u64-max sentinel (b128 vector stores).
// Pass 2 (scatter): per point, project; atomicMin_u64(ws[pixel],
//                   (float_bits(z)<<32)|index).  Positive-float bit ordering
//                   == unsigned ordering, so one near-memory atomic gives
//                   exactly (min z, tie-break min index).  The whole 78.6 MB
//                   z-buffer workspace is resident in the 192 MB L2.
// Pass 3 (gather):  per pixel, decode winner, recompute subpixel coords,
//                   write (xpix, ypix, z) or zeros.
//
// Streaming traffic (point loads, output stores) is non-temporal so it does
// not evict the L2-resident z-buffer.  N = 240*320 = 76800 = 300*256, so a
// 2D grid (300 blocks x nB batches) needs no div/mod and no tail checks.

namespace {

constexpr int   kB = 128;
constexpr int   kH = 240;
constexpr int   kW = 320;
constexpr int   kN = kH * kW;               // 76800 = 300 * 256
constexpr int   kTPB = 256;                 // 8 wave32 waves per block
constexpr int   kBlocksPerBatch = kN / kTPB;        // 300
constexpr float kFX = 294.68322709126955f;  // 589.3664541825391 * 0.5
constexpr float kFY = 294.68322709126955f;
constexpr float kCX = 160.25f;              // 320.5 * 0.5
constexpr float kCY = 120.25f;              // 240.5 * 0.5

// grid = (kN/2/kTPB, nB): each thread stores 16 B of sentinel.
__global__ void init_ws_kernel(ulonglong2* __restrict__ ws) {
  int i = blockIdx.x * kTPB + threadIdx.x;          // [0, kN/2)
  size_t off = (size_t)blockIdx.y * (kN / 2) + i;
  ws[off] = make_ulonglong2(~0ull, ~0ull);          // global_store_b128
}

// grid = (kBlocksPerBatch, nB)
__global__ void scatter_kernel(const float* __restrict__ pts,
                               unsigned long long* __restrict__ ws,
                               int baseB) {
  int i = blockIdx.x * kTPB + threadIdx.x;          // point id in batch
  int b = blockIdx.y;

  const float* p = pts + (size_t)(baseB + b) * (3 * kN);
  // Streamed once -> non-temporal loads keep L2 free for the z-buffer.
  float x = __builtin_nontemporal_load(p + i);
  float y = __builtin_nontemporal_load(p + kN + i);
  float z = __builtin_nontemporal_load(p + 2 * kN + i);

  if (!(z > 0.0f)) return;
  float xpix = (x / z) * kFX + kCX;
  float ypix = (y / z) * kFY + kCY;
  float rc = rintf(xpix);   // round-half-to-even, matches jnp.round
  float rr = rintf(ypix);
  if (!(rc >= 0.0f) || !(rc <= (float)(kW - 1)) ||
      !(rr >= 0.0f) || !(rr <= (float)(kH - 1)))
    return;
  int c = (int)rc;
  int r = (int)rr;

  // z > 0  =>  sign bit 0  =>  high word < 0x80000000 for every valid key.
  unsigned long long key =
      ((unsigned long long)__float_as_uint(z) << 32) | (unsigned int)i;
  atomicMin(ws + (size_t)b * kN + (r * kW + c), key);  // GLOBAL_ATOMIC_MIN_U64
}

// grid = (kBlocksPerBatch, nB)
__global__ void gather_kernel(const float* __restrict__ pts,
                              const unsigned long long* __restrict__ ws,
                              float* __restrict__ out, int baseB) {
  int i = blockIdx.x * kTPB + threadIdx.x;          // pixel id in batch
  int b = blockIdx.y;

  unsigned long long v = ws[(size_t)b * kN + i];    // hits L2
  unsigned int hi = (unsigned int)(v >> 32);

  float o0 = 0.0f, o1 = 0.0f, o2 = 0.0f;
  if (hi < 0x80000000u) {                           // a valid point won
    int idx = (int)(unsigned int)v;
    const float* p = pts + (size_t)(baseB + b) * (3 * kN);
    float x = p[idx];                               // gather; plane is L2-hot
    float y = p[kN + idx];
    float z = __uint_as_float(hi);                  // bit-exact winner depth
    o0 = (x / z) * kFX + kCX;
    o1 = (y / z) * kFY + kCY;
    o2 = z;
  }
  float* o = out + (size_t)(baseB + b) * (3 * kN);
  __builtin_nontemporal_store(o0, o + i);
  __builtin_nontemporal_store(o1, o + kN + i);
  __builtin_nontemporal_store(o2, o + 2 * kN + i);
}

}  // namespace

extern "C" void kernel_launch(void* const* d_in, const int* in_sizes, int n_in,
                              void* d_out, int out_size, void* d_ws, size_t ws_size,
                              hipStream_t stream) {
  const float* pts = (const float*)d_in[0];
  float* out = (float*)d_out;
  unsigned long long* ws = (unsigned long long*)d_ws;

  // Chunk batches by available scratch (8 B per pixel per batch).
  const size_t perBatch = (size_t)kN * sizeof(unsigned long long);
  size_t fit = perBatch ? (ws_size / perBatch) : 0;
  if (fit < 1) fit = 1;                 // assume harness provides >= 600 KB
  if (fit > (size_t)kB) fit = (size_t)kB;
  const int chunk = (int)fit;

  for (int b0 = 0; b0 < kB; b0 += chunk) {
    int nb = (kB - b0 < chunk) ? (kB - b0) : chunk;
    hipLaunchKernelGGL(init_ws_kernel,
                       dim3(kN / 2 / kTPB, nb), dim3(kTPB), 0, stream,
                       (ulonglong2*)ws);
    hipLaunchKernelGGL(scatter_kernel,
                       dim3(kBlocksPerBatch, nb), dim3(kTPB), 0, stream,
                       pts, ws, b0);
    hipLaunchKernelGGL(gather_kernel,
                       dim3(kBlocksPerBatch, nb), dim3(kTPB), 0, stream,
                       pts, ws, out, b0);
  }
}